// SSLGCNEncoder_39522289058399
// MI455X (gfx1250) — compile-verified
//
#include <hip/hip_runtime.h>

typedef __attribute__((ext_vector_type(2))) float v2f;
typedef __attribute__((ext_vector_type(8))) float v8f;

#define NNODES 50000
#define FIN    128
#define F1     256   // conv1 out
#define F2     128   // conv2 out
#define LN_EPS 1e-5f

struct alignas(16) EdgeRec { int s; int d; float norm; float pad; };

// ---------------- small elementwise kernels ----------------

__global__ void k_fill(float* __restrict__ p, float v, int n) {
  int i = blockIdx.x * blockDim.x + threadIdx.x;
  if (i < n) p[i] = v;
}

__global__ void k_deg(const int* __restrict__ dst, float* __restrict__ deg, int E) {
  int e = blockIdx.x * blockDim.x + threadIdx.x;
  if (e < E) atomicAdd(&deg[dst[e]], 1.0f);
}

__global__ void k_dinv(const float* __restrict__ deg, float* __restrict__ dinv, int n) {
  int i = blockIdx.x * blockDim.x + threadIdx.x;
  if (i < n) dinv[i] = rsqrtf(deg[i]);   // deg >= 1 always (self-loop)
}

__global__ void k_pack(const int* __restrict__ src, const int* __restrict__ dst,
                       const float* __restrict__ dinv, EdgeRec* __restrict__ pk, int E) {
  int e = blockIdx.x * blockDim.x + threadIdx.x;
  if (e >= E) return;
  EdgeRec r;
  r.s = src[e];
  r.d = dst[e];
  r.norm = dinv[r.s] * dinv[r.d];
  r.pad = 0.0f;
  pk[e] = r;
}

// out[i,:] = h[i,:] * dinv[i]^2  (+ bias)   -- self-loop term, float4-wide
template<int FQ>  // FQ = F/4
__global__ void k_self(const float4* __restrict__ h, const float* __restrict__ dinv,
                       const float4* __restrict__ bias, float4* __restrict__ out, int nq) {
  int idx = blockIdx.x * blockDim.x + threadIdx.x;
  if (idx >= nq) return;
  int i  = idx / FQ;
  int fq = idx % FQ;
  float s = dinv[i]; s = s * s;
  float4 v = h[idx];
  float4 o;
  o.x = v.x * s; o.y = v.y * s; o.z = v.z * s; o.w = v.w * s;
  if (bias) {
    float4 b = bias[fq];
    o.x += b.x; o.y += b.y; o.z += b.z; o.w += b.w;
  }
  out[idx] = o;
}

// edge scatter: out[dst,:] += h[src,:] * norm   (float4 load, 4x f32 atomics)
template<int FQ>  // FQ = F/4
__global__ void k_scatter(const float4* __restrict__ h, const EdgeRec* __restrict__ pk,
                          float* __restrict__ out, int total) {
  int tid = blockIdx.x * blockDim.x + threadIdx.x;
  if (tid >= total) return;
  int e  = tid / FQ;
  int fq = tid % FQ;
  EdgeRec er = pk[e];
  float4 v = h[(size_t)er.s * FQ + fq];
  float n = er.norm;
  float* o = out + ((size_t)er.d * FQ + fq) * 4;
  atomicAdd(o + 0, v.x * n);
  atomicAdd(o + 1, v.y * n);
  atomicAdd(o + 2, v.z * n);
  atomicAdd(o + 3, v.w * n);
}

// hn = LayerNorm(relu(agg + b)) * gamma + beta   -- one 256-thread block per row
__global__ void k_relu_ln(const float* __restrict__ agg, const float* __restrict__ b,
                          const float* __restrict__ g, const float* __restrict__ be,
                          float* __restrict__ out) {
  int row = blockIdx.x;
  int t   = threadIdx.x;            // 0..255
  float v = agg[(size_t)row * F1 + t] + b[t];
  v = fmaxf(v, 0.0f);
  __shared__ float red[F1];
  red[t] = v; __syncthreads();
  for (int s = F1 / 2; s > 0; s >>= 1) { if (t < s) red[t] += red[t + s]; __syncthreads(); }
  float mean = red[0] * (1.0f / F1);
  __syncthreads();
  float d = v - mean;
  red[t] = d * d; __syncthreads();
  for (int s = F1 / 2; s > 0; s >>= 1) { if (t < s) red[t] += red[t + s]; __syncthreads(); }
  float var = red[0] * (1.0f / F1);
  out[(size_t)row * F1 + t] = d * rsqrtf(var + LN_EPS) * g[t] + be[t];
}

// ---------------- fp32 WMMA GEMM: C[M,Nfull] = A[M,K] @ B[K,Nfull] ----------------
// One wave per (16-row strip, NT*16-column block). NT=8 -> 64 acc VGPRs.
// Per k-step: stage all NT B-fragments first (one wait covers 2*NT loads),
// then issue NT back-to-back WMMAs. B row pointer strength-reduced.
// Layout per CDNA5 ISA (16x16x4 f32):
//   A 16x4 : VGPR0 = K(0|2) by lane-half, VGPR1 = K(1|3); lane&15 = M
//   B 4x16 : mirrored, lane&15 = N column
//   C/D    : VGPR v, lane -> C[M = v + 8*(lane>>4), N = lane&15]
template<int NT>
__global__ __launch_bounds__(128, 1)
void k_gemm(const float* __restrict__ A, const float* __restrict__ B,
            float* __restrict__ C, int Mtiles, int K, int Nfull, int nb) {
  int wid = (blockIdx.x * blockDim.x + threadIdx.x) >> 5;  // wave32
  if (wid >= Mtiles * nb) return;                           // uniform per-wave exit
  int tile = wid / nb;            // M-strip
  int col0 = (wid % nb) * NT * 16;
  int lane = threadIdx.x & 31;
  int hl   = lane >> 4;           // lane-half
  int r    = lane & 15;

  const float* Ap = A + (size_t)(tile * 16 + r) * K + 2 * hl;   // a.x source
  const float* Bp = B + (size_t)(2 * hl) * Nfull + col0 + r;    // k-step row base

  v8f acc[NT];
  v8f zero = {};
#pragma unroll
  for (int nt = 0; nt < NT; ++nt) acc[nt] = zero;

  for (int k0 = 0; k0 < K; k0 += 4, Bp += 4 * Nfull) {
    v2f a;
    a.x = Ap[k0 + 0];
    a.y = Ap[k0 + 1];
    v2f bf[NT];
#pragma unroll
    for (int nt = 0; nt < NT; ++nt) {   // stage all B fragments: 2*NT loads in flight
      bf[nt].x = Bp[nt * 16];
      bf[nt].y = Bp[Nfull + nt * 16];
    }
#pragma unroll
    for (int nt = 0; nt < NT; ++nt) {   // then NT back-to-back WMMAs
      acc[nt] = __builtin_amdgcn_wmma_f32_16x16x4_f32(
          /*neg_a=*/false, a, /*neg_b=*/false, bf[nt],
          /*c_mod=*/(short)0, acc[nt], /*reuse_a=*/false, /*reuse_b=*/false);
    }
  }

  size_t base = (size_t)(tile * 16) * Nfull + col0;
#pragma unroll
  for (int nt = 0; nt < NT; ++nt)
#pragma unroll
    for (int v = 0; v < 8; ++v)
      C[base + (size_t)(v + 8 * hl) * Nfull + nt * 16 + r] = acc[nt][v];
}

// ---------------- launch ----------------

extern "C" void kernel_launch(void* const* d_in, const int* in_sizes, int n_in,
                              void* d_out, int out_size, void* d_ws, size_t ws_size,
                              hipStream_t stream) {
  const float* x     = (const float*)d_in[0];
  const int*   ei    = (const int*)  d_in[1];
  const float* W1    = (const float*)d_in[2];
  const float* b1    = (const float*)d_in[3];
  const float* gamma = (const float*)d_in[4];
  const float* beta  = (const float*)d_in[5];
  const float* W2    = (const float*)d_in[6];
  const float* b2    = (const float*)d_in[7];

  const int N = NNODES;
  const int E = in_sizes[1] / 2;
  const int* srcI = ei;
  const int* dstI = ei + E;

  // workspace carve (hn aliases h1, h2 aliases agg1)
  float* h1   = (float*)d_ws;                 // N*F1
  float* agg1 = h1 + (size_t)N * F1;          // N*F1
  float* deg  = agg1 + (size_t)N * F1;        // N
  float* dinv = deg + N;                      // N
  uintptr_t pka = ((uintptr_t)(dinv + N) + 15) & ~(uintptr_t)15;
  EdgeRec* pack = (EdgeRec*)pka;              // E * 16B
  float* hn = h1;
  float* h2 = agg1;
  float* out = (float*)d_out;

  const int T = 256;
  const int Mtiles = N / 16;                  // 3125

  // degree (self-loop contributes 1) -> dinv -> packed edge records
  k_fill<<<(N + T - 1) / T, T, 0, stream>>>(deg, 1.0f, N);
  k_deg <<<(E + T - 1) / T, T, 0, stream>>>(dstI, deg, E);
  k_dinv<<<(N + T - 1) / T, T, 0, stream>>>(deg, dinv, N);
  k_pack<<<(E + T - 1) / T, T, 0, stream>>>(srcI, dstI, dinv, pack, E);

  // conv1: h1 = x @ W1   (K=128, N=256 -> 2 column-blocks of 128)
  {
    int waves = Mtiles * 2;
    k_gemm<8><<<(waves + 3) / 4, 128, 0, stream>>>(x, W1, h1, Mtiles, FIN, F1, 2);
  }
  // agg1 = self-loop term, then scatter edges
  {
    int nq = N * (F1 / 4);
    k_self<F1 / 4><<<(nq + T - 1) / T, T, 0, stream>>>(
        (const float4*)h1, dinv, nullptr, (float4*)agg1, nq);
    int total = E * (F1 / 4);
    k_scatter<F1 / 4><<<(total + T - 1) / T, T, 0, stream>>>(
        (const float4*)h1, pack, agg1, total);
  }
  // hn = LN(relu(agg1 + b1))
  k_relu_ln<<<N, F1, 0, stream>>>(agg1, b1, gamma, beta, hn);

  // conv2: h2 = hn @ W2   (K=256, N=128 -> 1 column-block)
  {
    int waves = Mtiles;
    k_gemm<8><<<(waves + 3) / 4, 128, 0, stream>>>(hn, W2, h2, Mtiles, F1, F2, 1);
  }
  // out = self-loop term + b2, then scatter edges
  {
    int nq = N * (F2 / 4);
    k_self<F2 / 4><<<(nq + T - 1) / T, T, 0, stream>>>(
        (const float4*)h2, dinv, (const float4*)b2, (float4*)out, nq);
    int total = E * (F2 / 4);
    k_scatter<F2 / 4><<<(total + T - 1) / T, T, 0, stream>>>(
        (const float4*)h2, pack, out, total);
  }
}